// Predictor_13408887898428
// MI455X (gfx1250) — compile-verified
//
#include <hip/hip_runtime.h>
#include <math.h>

// ---------------------------------------------------------------------------
// DTI predictor forward pass for gfx1250 (MI455X), fp32 via V_WMMA_F32_16X16X4_F32.
// Algebraic optimization: atten.mean(axis) @ W commutes -> pairwise relu means
// computed elementwise (0.25 GFLOP) instead of 9.8 GFLOP GEMM + 307 MB tensor.
// Codegen optimization vs previous round: branch-free inner loops (row-clamped
// pointers + load-then-select for the single K tail step), vectorized v2f
// (global_load_b64) fragment loads. Scratch: ~54 MB of d_ws.
// ---------------------------------------------------------------------------

typedef __attribute__((ext_vector_type(2))) float v2f;
typedef __attribute__((ext_vector_type(8))) float v8f;

#define BB    8
#define NATOM 50
#define ATOMF 34
#define LSMI  100
#define LPRO  1000
#define HH    64
#define L1T   150   // LSMI + NATOM
#define LCS   112   // smi length padded to 16
#define LCP   1008  // pro length padded to 16

__device__ __forceinline__ float wave_max(float v){
  #pragma unroll
  for(int m=16;m>=1;m>>=1) v = fmaxf(v, __shfl_xor(v,m,32));
  return v;
}
__device__ __forceinline__ float wave_sum(float v){
  #pragma unroll
  for(int m=16;m>=1;m>>=1) v += __shfl_xor(v,m,32);
  return v;
}

// ---------------------------------------------------------------------------
// Generic fp32 WMMA GEMM: out[M,N] = act( X[M,K](row stride lda) @ W[K,N]
//                                          + bias[N] + res[M,N] )
// One wave per 16x16 tile; batched via blockIdx.z strides. N must be a
// multiple of 16. Out-of-range rows are handled by CLAMPING the A row pointer
// (D row m depends only on A row m, and invalid D rows are never stored), so
// the main loop is completely branch-free: one b64 A load + two b32 B loads
// + one v_wmma per K-step. K%4 handled in a single uniform-guarded tail step
// with load-then-select (no divergent predicated loads).
// ACT: 0=none, 1=relu, 2=sigmoid
// ---------------------------------------------------------------------------
template<int ACT>
__global__ void __launch_bounds__(32) k_gemm(
    const float* __restrict__ X, long sX, int lda,
    const float* __restrict__ W, long sW,
    const float* __restrict__ bias,
    const float* __restrict__ res, long sRes,
    float* __restrict__ out, long sOut,
    int M, int N, int K)
{
  X   += (long)blockIdx.z * sX;
  W   += (long)blockIdx.z * sW;
  out += (long)blockIdx.z * sOut;
  if (res) res += (long)blockIdx.z * sRes;

  const int m0   = blockIdx.x * 16;
  const int n0   = blockIdx.y * 16;
  const int lane = threadIdx.x;
  const int half = lane >> 4;      // 0: k={0,1}, 1: k={2,3}
  const int l16  = lane & 15;
  const int arow = m0 + l16;
  const int bcol = n0 + l16;
  const int arc  = (arow < M) ? arow : (M - 1);   // clamped: always valid

  const float* __restrict__ Xr = X + (long)arc * lda + half * 2;
  const float* __restrict__ Wc = W + (long)(half * 2) * N + bcol;

  v8f acc = {};
  const int kk4 = K >> 2;
  for (int kk = 0; kk < kk4; ++kk) {
    v2f a = *(const v2f*)(Xr + kk * 4);
    v2f b;
    b.x = Wc[(long)(kk * 4) * N];
    b.y = Wc[(long)(kk * 4) * N + N];
    acc = __builtin_amdgcn_wmma_f32_16x16x4_f32(false, a, false, b,
                                                (short)0, acc, false, false);
  }
  if (K & 3) {   // uniform scalar branch: EXEC stays all-1s
    const int k0 = (kk4 << 2) + half * 2;
    const int ka = (k0     < K) ? k0     : (K - 1);
    const int kb = (k0 + 1 < K) ? k0 + 1 : (K - 1);
    const float* __restrict__ Xrow = X + (long)arc * lda;
    const float ax = Xrow[ka],               ay = Xrow[kb];
    const float bx = W[(long)ka * N + bcol], by = W[(long)kb * N + bcol];
    v2f a, b;
    a.x = (k0     < K) ? ax : 0.f;
    a.y = (k0 + 1 < K) ? ay : 0.f;
    b.x = (k0     < K) ? bx : 0.f;
    b.y = (k0 + 1 < K) ? by : 0.f;
    acc = __builtin_amdgcn_wmma_f32_16x16x4_f32(false, a, false, b,
                                                (short)0, acc, false, false);
  }

  #pragma unroll
  for (int v = 0; v < 8; ++v) {
    const int row = m0 + v + half * 8;
    if (row < M) {
      float r = acc[v];
      if (bias) r += bias[bcol];
      if (res)  r += res[(long)row * N + bcol];
      if (ACT == 1)      r = fmaxf(r, 0.f);
      else if (ACT == 2) r = 1.f / (1.f + expf(-r));
      out[(long)row * N + bcol] = r;
    }
  }
}

// ---------------------------------------------------------------------------
// Attention scores: S[b, :Lc, :Lc] = Q[b]·K[b]^T * scale. Padded rows/cols get
// -1e30 at the store (so softmax -> exactly 0 and S@V may stop at K=L), which
// lets us CLAMP both Q-row and K-col pointers: inner loop is two b64 loads +
// one v_wmma, fully unrolled over K=64.
// ---------------------------------------------------------------------------
__global__ void __launch_bounds__(32) k_scores(
    const float* __restrict__ Q, const float* __restrict__ Km,
    float* __restrict__ S, int L, int Lc, float scale)
{
  Q  += (long)blockIdx.z * L * HH;
  Km += (long)blockIdx.z * L * HH;
  S  += (long)blockIdx.z * Lc * Lc;

  const int r0 = blockIdx.x * 16, c0 = blockIdx.y * 16;
  const int lane = threadIdx.x, half = lane >> 4, l16 = lane & 15;
  const int qrow = r0 + l16, kcol = c0 + l16;
  const int qr = (qrow < L) ? qrow : (L - 1);
  const int kc = (kcol < L) ? kcol : (L - 1);
  const float* __restrict__ Qr = Q  + (long)qr * HH + half * 2;
  const float* __restrict__ Kr = Km + (long)kc * HH + half * 2;

  v8f acc = {};
  #pragma unroll
  for (int kk = 0; kk < 16; ++kk) {   // K-dim = 64
    v2f a = *(const v2f*)(Qr + kk * 4);
    v2f b = *(const v2f*)(Kr + kk * 4);
    acc = __builtin_amdgcn_wmma_f32_16x16x4_f32(false, a, false, b,
                                                (short)0, acc, false, false);
  }
  #pragma unroll
  for (int v = 0; v < 8; ++v) {
    const int row = r0 + v + half * 8;
    const float val = (row < L && kcol < L) ? acc[v] * scale : -1e30f;
    S[(long)row * Lc + kcol] = val;
  }
}

// Row softmax, one wave per row of length Lc (rows laid out [B*Lc, Lc]).
__global__ void __launch_bounds__(32) k_softmax(float* __restrict__ S, int Lc)
{
  float* row = S + (long)blockIdx.x * Lc;
  const int t = threadIdx.x;
  float m = -3.0e38f;
  for (int c = t; c < Lc; c += 32) m = fmaxf(m, row[c]);
  m = wave_max(m);
  float s = 0.f;
  for (int c = t; c < Lc; c += 32) s += expf(row[c] - m);
  s = wave_sum(s);
  const float inv = 1.f / s;
  for (int c = t; c < Lc; c += 32) row[c] = expf(row[c] - m) * inv;
}

// Embedding gather: out[r,:] = table[ids[r],:]  (row 0 of table is zeros).
__global__ void k_embed(const int* __restrict__ ids, const float* __restrict__ tab,
                        float* __restrict__ out)
{
  out[(long)blockIdx.x * HH + threadIdx.x] =
      tab[(long)ids[blockIdx.x] * HH + threadIdx.x];
}

// GAT per-node scores: s1 = h·a[:64], s2 = h·a[64:], one wave per node row.
__global__ void __launch_bounds__(32) k_gat_sc(
    const float* __restrict__ h, const float* __restrict__ ga,
    float* __restrict__ s1, float* __restrict__ s2)
{
  const int r = blockIdx.x, t = threadIdx.x;
  const float h0 = h[(long)r * HH + t], h1 = h[(long)r * HH + t + 32];
  float a = h0 * ga[t]      + h1 * ga[t + 32];
  float b = h0 * ga[64 + t] + h1 * ga[96 + t];
  a = wave_sum(a); b = wave_sum(b);
  if (t == 0) { s1[r] = a; s2[r] = b; }
}

// GAT attention row: softmax over masked leaky_relu(s1_i + s2_j), then
// drug_gat[b,i,:] = elu(sum_j alpha_j * h[b,j,:]).  One 64-thread block per (b,i).
__global__ void k_gat_attn(const float* __restrict__ s1, const float* __restrict__ s2,
                           const int* __restrict__ adj, const float* __restrict__ h,
                           float* __restrict__ dg)
{
  const int b = blockIdx.x / NATOM, i = blockIdx.x % NATOM, t = threadIdx.x;
  __shared__ float al[NATOM];
  __shared__ float inv_s;
  if (t < NATOM) {
    float ev = s1[b * NATOM + i] + s2[b * NATOM + t];
    ev = ev > 0.f ? ev : 0.2f * ev;                       // leaky_relu(0.2)
    const bool msk = (adj[((long)b * NATOM + i) * NATOM + t] > 0) || (t == i);
    al[t] = msk ? ev : -1e9f;
  }
  __syncthreads();
  if (t == 0) {
    float mx = -3.0e38f;
    for (int j = 0; j < NATOM; ++j) mx = fmaxf(mx, al[j]);
    float s = 0.f;
    for (int j = 0; j < NATOM; ++j) { al[j] = expf(al[j] - mx); s += al[j]; }
    inv_s = 1.f / s;
  }
  __syncthreads();
  float o = 0.f;
  for (int j = 0; j < NATOM; ++j)
    o += al[j] * h[((long)b * NATOM + j) * HH + t];
  o *= inv_s;
  dg[((long)b * NATOM + i) * HH + t] = o > 0.f ? o : (expf(o) - 1.f);  // elu
}

// smi_attss = concat([smi_att, drug_gat], axis=1)
__global__ void k_cat_attss(const float* __restrict__ sa, const float* __restrict__ dg,
                            float* __restrict__ out)
{
  const int b = blockIdx.x / L1T, l = blockIdx.x % L1T, c = threadIdx.x;
  const float v = (l < LSMI)
      ? sa[((long)b * LSMI + l) * HH + c]
      : dg[((long)b * NATOM + (l - LSMI)) * HH + c];
  out[(long)blockIdx.x * HH + c] = v;
}

// M_s[b,i,c] = mean_j relu(smi_attss[b,i,c] + pro_att[b,j,c])
__global__ void k_pair_ms(const float* __restrict__ attss, const float* __restrict__ patt,
                          float* __restrict__ Ms)
{
  const int b = blockIdx.x / L1T, i = blockIdx.x % L1T, c = threadIdx.x;
  const float s = attss[((long)b * L1T + i) * HH + c];
  const float* __restrict__ p = patt + (long)b * LPRO * HH + c;
  float acc = 0.f;
  for (int j = 0; j < LPRO; ++j) acc += fmaxf(s + p[(long)j * HH], 0.f);
  Ms[((long)b * L1T + i) * HH + c] = acc * (1.f / LPRO);
}

// M_p[b,j,c] = mean_i relu(smi_attss[b,i,c] + pro_att[b,j,c])
__global__ void k_pair_mp(const float* __restrict__ attss, const float* __restrict__ patt,
                          float* __restrict__ Mp)
{
  const int b = blockIdx.x / LPRO, j = blockIdx.x % LPRO, c = threadIdx.x;
  const float pv = patt[((long)b * LPRO + j) * HH + c];
  const float* __restrict__ s = attss + (long)b * L1T * HH + c;
  float acc = 0.f;
  for (int i = 0; i < L1T; ++i) acc += fmaxf(pv + s[(long)i * HH], 0.f);
  Mp[((long)b * LPRO + j) * HH + c] = acc * (1.f / L1T);
}

// vec[b,c] = mean_rows (0.5*base + base*gate)
__global__ void k_mix(const float* __restrict__ base, const float* __restrict__ gate,
                      float* __restrict__ out, int rows)
{
  const int b = blockIdx.x, c = threadIdx.x;
  float acc = 0.f;
  for (int r = 0; r < rows; ++r) {
    const float v = base[((long)b * rows + r) * HH + c];
    const float g = gate[((long)b * rows + r) * HH + c];
    acc += 0.5f * v + v * g;
  }
  out[b * HH + c] = acc / (float)rows;
}

__global__ void k_cat_x(const float* __restrict__ s, const float* __restrict__ p,
                        float* __restrict__ x)
{
  const int b = blockIdx.x, c = threadIdx.x;
  x[b * 128 + c] = (c < 64) ? s[b * 64 + c] : p[b * 64 + (c - 64)];
}

// Final tiny layer: out[8,2] = x3[8,512] @ w[512,2] + b[2]
__global__ void k_o4(const float* __restrict__ x3, const float* __restrict__ w,
                     const float* __restrict__ bi, float* __restrict__ out)
{
  const int t = threadIdx.x;
  if (t >= 16) return;
  const int b = t >> 1, n = t & 1;
  float acc = bi[n];
  for (int k = 0; k < 512; ++k) acc += x3[b * 512 + k] * w[k * 2 + n];
  out[b * 2 + n] = acc;
}

// ---------------------------------------------------------------------------
extern "C" void kernel_launch(void* const* d_in, const int* in_sizes, int n_in,
                              void* d_out, int out_size, void* d_ws, size_t ws_size,
                              hipStream_t stream)
{
  const float* compound = (const float*)d_in[0];
  const int*   adj      = (const int*)  d_in[1];
  const int*   smi_ids  = (const int*)  d_in[2];
  const int*   prot_ids = (const int*)  d_in[3];
  const float* prot_emb = (const float*)d_in[4];
  const float* smi_emb  = (const float*)d_in[5];
  const float* smi_wq   = (const float*)d_in[6];
  const float* smi_wk   = (const float*)d_in[7];
  const float* smi_wv   = (const float*)d_in[8];
  const float* smi_wo   = (const float*)d_in[9];
  const float* pro_wq   = (const float*)d_in[10];
  const float* pro_wk   = (const float*)d_in[11];
  const float* pro_wv   = (const float*)d_in[12];
  const float* pro_wo   = (const float*)d_in[13];
  const float* att_w    = (const float*)d_in[14];
  const float* att_b    = (const float*)d_in[15];
  const float* gat_w    = (const float*)d_in[16];
  const float* gat_a    = (const float*)d_in[17];
  const float* o1_w = (const float*)d_in[18]; const float* o1_b = (const float*)d_in[19];
  const float* o2_w = (const float*)d_in[20]; const float* o2_b = (const float*)d_in[21];
  const float* o3_w = (const float*)d_in[22]; const float* o3_b = (const float*)d_in[23];
  const float* o4_w = (const float*)d_in[24]; const float* o4_b = (const float*)d_in[25];
  float* outp = (float*)d_out;

  // -- scratch carve-out (floats) --
  float* ws = (float*)d_ws;
  size_t off = 0;
  auto alloc = [&](size_t n) { float* p = ws + off; off += n; return p; };
  float* h        = alloc((size_t)BB * NATOM * HH);
  float* s1       = alloc((size_t)BB * NATOM);
  float* s2       = alloc((size_t)BB * NATOM);
  float* drug_gat = alloc((size_t)BB * NATOM * HH);
  float* se       = alloc((size_t)BB * LSMI * HH);
  float* pe       = alloc((size_t)BB * LPRO * HH);
  float* sq       = alloc((size_t)BB * LSMI * HH);
  float* sk       = alloc((size_t)BB * LSMI * HH);
  float* sv       = alloc((size_t)BB * LSMI * HH);
  float* Ssmi     = alloc((size_t)BB * LCS * LCS);
  float* so       = alloc((size_t)BB * LSMI * HH);
  float* smi_tf   = alloc((size_t)BB * LSMI * HH);
  float* pq       = alloc((size_t)BB * LPRO * HH);
  float* pk       = alloc((size_t)BB * LPRO * HH);
  float* pv       = alloc((size_t)BB * LPRO * HH);
  float* Spro     = alloc((size_t)BB * LCP * LCP);
  float* po       = alloc((size_t)BB * LPRO * HH);
  float* pro_tf   = alloc((size_t)BB * LPRO * HH);
  float* smi_att  = alloc((size_t)BB * LSMI * HH);
  float* pro_att  = alloc((size_t)BB * LPRO * HH);
  float* attss    = alloc((size_t)BB * L1T * HH);
  float* Ms       = alloc((size_t)BB * L1T * HH);
  float* Mp       = alloc((size_t)BB * LPRO * HH);
  float* smi_atts = alloc((size_t)BB * L1T * HH);
  float* pro_atts = alloc((size_t)BB * LPRO * HH);
  float* smi_vec  = alloc((size_t)BB * HH);
  float* pro_vec  = alloc((size_t)BB * HH);
  float* xcat     = alloc((size_t)BB * 128);
  float* x1       = alloc((size_t)BB * 1024);
  float* x2       = alloc((size_t)BB * 1024);
  float* x3       = alloc((size_t)BB * 512);
  (void)off; (void)ws_size; (void)in_sizes; (void)n_in; (void)out_size;

  #define GEMM(ACT, X,sX,lda, W,sW, BIAS, RES,sRES, OUT,sOUT, M,N,K, Z) \
    k_gemm<ACT><<<dim3(((M)+15)/16,(N)/16,(Z)), dim3(32), 0, stream>>>( \
      X,(long)(sX),lda, W,(long)(sW), BIAS, RES,(long)(sRES), OUT,(long)(sOUT), M,N,K)

  // 1. embeddings
  k_embed<<<dim3(BB*LSMI), dim3(HH), 0, stream>>>(smi_ids, smi_emb, se);
  k_embed<<<dim3(BB*LPRO), dim3(HH), 0, stream>>>(prot_ids, prot_emb, pe);

  // 2. GAT
  GEMM(0, compound,0,ATOMF, gat_w,0, nullptr, nullptr,0, h,0, BB*NATOM,HH,ATOMF, 1);
  k_gat_sc<<<dim3(BB*NATOM), dim3(32), 0, stream>>>(h, gat_a, s1, s2);
  k_gat_attn<<<dim3(BB*NATOM), dim3(HH), 0, stream>>>(s1, s2, adj, h, drug_gat);

  // 3. SMILES transformer
  GEMM(0, se,0,HH, smi_wq,0, nullptr, nullptr,0, sq,0, BB*LSMI,HH,HH, 1);
  GEMM(0, se,0,HH, smi_wk,0, nullptr, nullptr,0, sk,0, BB*LSMI,HH,HH, 1);
  GEMM(0, se,0,HH, smi_wv,0, nullptr, nullptr,0, sv,0, BB*LSMI,HH,HH, 1);
  k_scores<<<dim3(LCS/16, LCS/16, BB), dim3(32), 0, stream>>>(sq, sk, Ssmi, LSMI, LCS, 0.125f);
  k_softmax<<<dim3(BB*LCS), dim3(32), 0, stream>>>(Ssmi, LCS);
  GEMM(0, Ssmi,(size_t)LCS*LCS,LCS, sv,(size_t)LSMI*HH, nullptr, nullptr,0,
       so,(size_t)LSMI*HH, LSMI,HH,LSMI, BB);
  GEMM(0, so,0,HH, smi_wo,0, nullptr, se,0, smi_tf,0, BB*LSMI,HH,HH, 1);

  // 4. Protein transformer
  GEMM(0, pe,0,HH, pro_wq,0, nullptr, nullptr,0, pq,0, BB*LPRO,HH,HH, 1);
  GEMM(0, pe,0,HH, pro_wk,0, nullptr, nullptr,0, pk,0, BB*LPRO,HH,HH, 1);
  GEMM(0, pe,0,HH, pro_wv,0, nullptr, nullptr,0, pv,0, BB*LPRO,HH,HH, 1);
  k_scores<<<dim3(LCP/16, LCP/16, BB), dim3(32), 0, stream>>>(pq, pk, Spro, LPRO, LCP, 0.125f);
  k_softmax<<<dim3(BB*LCP), dim3(32), 0, stream>>>(Spro, LCP);
  GEMM(0, Spro,(size_t)LCP*LCP,LCP, pv,(size_t)LPRO*HH, nullptr, nullptr,0,
       po,(size_t)LPRO*HH, LPRO,HH,LPRO, BB);
  GEMM(0, po,0,HH, pro_wo,0, nullptr, pe,0, pro_tf,0, BB*LPRO,HH,HH, 1);

  // 5. cross-attention projections
  GEMM(0, smi_tf,0,HH, att_w,0, att_b, nullptr,0, smi_att,0, BB*LSMI,HH,HH, 1);
  GEMM(0, pro_tf,0,HH, att_w,0, att_b, nullptr,0, pro_att,0, BB*LPRO,HH,HH, 1);
  k_cat_attss<<<dim3(BB*L1T), dim3(HH), 0, stream>>>(smi_att, drug_gat, attss);

  // 6. pairwise relu means (mean commuted through the linear att_w projection)
  k_pair_ms<<<dim3(BB*L1T),  dim3(HH), 0, stream>>>(attss, pro_att, Ms);
  k_pair_mp<<<dim3(BB*LPRO), dim3(HH), 0, stream>>>(attss, pro_att, Mp);
  GEMM(2, Ms,0,HH, att_w,0, att_b, nullptr,0, smi_atts,0, BB*L1T,HH,HH, 1);
  GEMM(2, Mp,0,HH, att_w,0, att_b, nullptr,0, pro_atts,0, BB*LPRO,HH,HH, 1);

  // 7. gated means + concat
  k_mix<<<dim3(BB), dim3(HH), 0, stream>>>(attss,  smi_atts, smi_vec, L1T);
  k_mix<<<dim3(BB), dim3(HH), 0, stream>>>(pro_tf, pro_atts, pro_vec, LPRO);
  k_cat_x<<<dim3(BB), dim3(128), 0, stream>>>(smi_vec, pro_vec, xcat);

  // 8. MLP head
  GEMM(1, xcat,0,128, o1_w,0, o1_b, nullptr,0, x1,0, BB,1024,128, 1);
  GEMM(1, x1,0,1024,  o2_w,0, o2_b, nullptr,0, x2,0, BB,1024,1024, 1);
  GEMM(1, x2,0,1024,  o3_w,0, o3_b, nullptr,0, x3,0, BB,512,1024, 1);
  k_o4<<<dim3(1), dim3(32), 0, stream>>>(x3, o4_w, o4_b, outp);

  #undef GEMM
}